// Rot2D_10299331576024
// MI455X (gfx1250) — compile-verified
//
#include <hip/hip_runtime.h>
#include <math.h>

// Problem constants (from reference): x:(32,256,256,16) f32, W:(256,256,16) f32
#define Bn 32
#define Hn 256
#define Wn 256
#define Cn 16
#define KDIM (Wn * Cn)        // 4096 contiguous floats per (b,h) row of x
#define MROWS (Bn * Hn)       // 8192 rows
#define NTILES (MROWS / 16)   // 512 WMMA row-tiles (16 tiles per batch)

typedef __attribute__((ext_vector_type(2))) float v2f;
typedef __attribute__((ext_vector_type(8))) float v8f;

// ---------------------------------------------------------------------------
// Stage 0: Wsum[w,c] = sum_k W[w,k,c]   (v[j], j = w*16 + c, 4096 floats)
// ---------------------------------------------------------------------------
__global__ void Rot2D_wsum_kernel(const float* __restrict__ Wm,
                                  float* __restrict__ v) {
    int j = blockIdx.x * blockDim.x + threadIdx.x;   // 0..4095
    int w = j >> 4;
    int c = j & 15;
    const float* p = Wm + (size_t)w * (size_t)KDIM + c;
    float s = 0.0f;
#pragma unroll 8
    for (int k = 0; k < 256; ++k) s += p[k * Cn];
    v[j] = s;
}

// ---------------------------------------------------------------------------
// Stage 1: per-tile partial dot products via V_WMMA_F32_16X16X4_F32.
// One wave handles one 16-row tile (rows are (b,h) pairs; 256 rows per b, so
// a tile never crosses batches). D = A(16x4) x B(4x16) + C accumulated over
// K=4096 in steps of 4. B columns all hold the same Wsum values, so every D
// column carries the 16 row sums; lanes 0 / 16 own rows 0-7 / 8-15.
// Deterministic: fixed accumulation order per wave, fixed tile->partial slot.
// ---------------------------------------------------------------------------
__global__ void Rot2D_dot_kernel(const float* __restrict__ x,
                                 const float* __restrict__ v,
                                 float* __restrict__ partial) {
    int tile = (blockIdx.x * blockDim.x + threadIdx.x) >> 5;  // global wave id
    int lane = threadIdx.x & 31;
    int half = lane >> 4;      // 0: K={0,1}, 1: K={2,3} (A layout, ISA 7.12.2)
    int mrow = lane & 15;

    const float* rowp = x + (size_t)(tile * 16 + mrow) * (size_t)KDIM + 2 * half;
    const float* vp   = v + 2 * half;

    float tot;
#if __has_builtin(__builtin_amdgcn_wmma_f32_16x16x4_f32)
    v8f acc = {};
#pragma unroll 4
    for (int k0 = 0; k0 < KDIM; k0 += 4) {
        v2f a = *(const v2f*)(rowp + k0);   // A: row 'mrow', K = k0+2*half+{0,1}
        v2f b = *(const v2f*)(vp + k0);     // B: broadcast Wsum across N
        acc = __builtin_amdgcn_wmma_f32_16x16x4_f32(
            false, a, false, b, (short)0, acc, false, false);
    }
    // lanes 0-15 hold rows 0-7 in acc[0..7]; lanes 16-31 hold rows 8-15
    float s = acc[0] + acc[1] + acc[2] + acc[3] +
              acc[4] + acc[5] + acc[6] + acc[7];
    tot = __shfl(s, 0, 32) + __shfl(s, 16, 32);
#else
    // Deterministic VALU fallback: same data assignment, full-wave tree sum
    float s = 0.0f;
#pragma unroll 4
    for (int k0 = 0; k0 < KDIM; k0 += 4) {
        v2f a = *(const v2f*)(rowp + k0);
        v2f b = *(const v2f*)(vp + k0);
        s += a.x * b.x + a.y * b.y;
    }
    for (int off = 16; off > 0; off >>= 1) s += __shfl_down(s, off, 32);
    tot = __shfl(s, 0, 32);
#endif
    if (lane == 0) partial[tile] = tot;
}

// ---------------------------------------------------------------------------
// Stage 2: angle[b] = fixed-order sum of 16 tile partials; then
// theta = deg2rad(floor(rad2deg(angle) mod 360)); store cos/sin per batch.
// ---------------------------------------------------------------------------
__global__ void Rot2D_angle_kernel(const float* __restrict__ partial,
                                   float* __restrict__ trig) {
    int b = threadIdx.x;
    if (b >= Bn) return;
    float a = 0.0f;
#pragma unroll
    for (int t = 0; t < NTILES / Bn; ++t) a += partial[b * (NTILES / Bn) + t];
    float deg = a * 57.29577951308232f;           // rad2deg (f32, as reference)
    float m = fmodf(deg, 360.0f);
    if (m < 0.0f) m += 360.0f;                    // jnp mod: sign of divisor
    m = floorf(m);
    float th = m * 0.017453292519943295f;         // deg2rad
    trig[2 * b + 0] = cosf(th);
    trig[2 * b + 1] = sinf(th);
}

// ---------------------------------------------------------------------------
// Stage 3: inverse-map bilinear rotation about (128,128), zero border.
// One thread per output pixel (16 channels = 4 x float4). Coalesced stores;
// neighboring threads gather along a rotated line -> good L0/L2 locality.
// ---------------------------------------------------------------------------
__global__ void Rot2D_rotate_kernel(const float* __restrict__ x,
                                    const float* __restrict__ trig,
                                    float* __restrict__ out) {
    int idx = blockIdx.x * blockDim.x + threadIdx.x;   // 0 .. B*H*W-1
    int c = idx & (Wn - 1);
    int r = (idx >> 8) & (Hn - 1);
    int b = idx >> 16;

    float cs = trig[2 * b + 0];
    float sn = trig[2 * b + 1];

    float fx = (float)(c - 128);
    float fy = (float)(r - 128);
    float xs = cs * fx - sn * fy + 128.0f;   // src col
    float ys = sn * fx + cs * fy + 128.0f;   // src row
    float x0f = floorf(xs), y0f = floorf(ys);
    float tx = xs - x0f, ty = ys - y0f;
    int x0 = (int)x0f, y0 = (int)y0f;

    float w00 = (1.0f - tx) * (1.0f - ty);
    float w01 = tx * (1.0f - ty);
    float w10 = (1.0f - tx) * ty;
    float w11 = tx * ty;

    bool g00 = ((unsigned)x0 < 256u)       && ((unsigned)y0 < 256u);
    bool g01 = ((unsigned)(x0 + 1) < 256u) && ((unsigned)y0 < 256u);
    bool g10 = ((unsigned)x0 < 256u)       && ((unsigned)(y0 + 1) < 256u);
    bool g11 = ((unsigned)(x0 + 1) < 256u) && ((unsigned)(y0 + 1) < 256u);

    const float* img = x + (size_t)b * (size_t)(Hn * Wn * Cn);
    const float4* p00 = (const float4*)(img + ((size_t)y0 * Wn + x0) * Cn);
    const float4* p01 = (const float4*)(img + ((size_t)y0 * Wn + (x0 + 1)) * Cn);
    const float4* p10 = (const float4*)(img + ((size_t)(y0 + 1) * Wn + x0) * Cn);
    const float4* p11 = (const float4*)(img + ((size_t)(y0 + 1) * Wn + (x0 + 1)) * Cn);

    float4* o = (float4*)(out + (size_t)idx * Cn);
    const float4 z = make_float4(0.f, 0.f, 0.f, 0.f);

#pragma unroll
    for (int q = 0; q < 4; ++q) {
        float4 a00 = g00 ? p00[q] : z;
        float4 a01 = g01 ? p01[q] : z;
        float4 a10 = g10 ? p10[q] : z;
        float4 a11 = g11 ? p11[q] : z;
        float4 res;
        res.x = a00.x * w00 + a01.x * w01 + a10.x * w10 + a11.x * w11;
        res.y = a00.y * w00 + a01.y * w01 + a10.y * w10 + a11.y * w11;
        res.z = a00.z * w00 + a01.z * w01 + a10.z * w10 + a11.z * w11;
        res.w = a00.w * w00 + a01.w * w01 + a10.w * w10 + a11.w * w11;
        o[q] = res;
    }
}

// ---------------------------------------------------------------------------
// Workspace layout (floats): [0,4096) Wsum | [4096,4608) tile partials |
// [4608,4672) cos/sin pairs. Needs ~18.7 KB of d_ws.
// ---------------------------------------------------------------------------
extern "C" void kernel_launch(void* const* d_in, const int* in_sizes, int n_in,
                              void* d_out, int out_size, void* d_ws, size_t ws_size,
                              hipStream_t stream) {
    const float* x  = (const float*)d_in[0];   // (32,256,256,16) f32
    const float* Wm = (const float*)d_in[1];   // (256,256,16)    f32
    float* out = (float*)d_out;                // (32,256,256,16) f32
    float* ws  = (float*)d_ws;

    float* v       = ws;            // 4096
    float* partial = ws + 4096;     // 512
    float* trig    = ws + 4608;     // 64

    Rot2D_wsum_kernel  <<<KDIM / 256, 256, 0, stream>>>(Wm, v);
    Rot2D_dot_kernel   <<<NTILES / 8, 256, 0, stream>>>(x, v, partial);
    Rot2D_angle_kernel <<<1, 32, 0, stream>>>(partial, trig);
    Rot2D_rotate_kernel<<<(Bn * Hn * Wn) / 256, 256, 0, stream>>>(x, trig, out);
}